// TruncatedLIF_56118042689792
// MI455X (gfx1250) — compile-verified
//
#include <hip/hip_runtime.h>
#include <stdint.h>

// ---------------- problem constants (fixed by the reference) ----------------
#define B_DIM   8
#define T_DIM   4096
#define D_DIM   1024
#define K_TRUNC 8

// ---------------- tiling ----------------
#define THREADS 256          // 8 waves (wave32)
#define CH      4            // channels per thread -> 256*4 = 1024 = D
#define TCHUNK  128          // time rows per workgroup (halo = 8 rows recomputed)
#define TROWS   8            // rows per TDM tile -> 8*1024*4B = 32 KB
#define NTILES  (TCHUNK / TROWS)

typedef uint32_t v4u __attribute__((ext_vector_type(4)));
typedef int      v8i __attribute__((ext_vector_type(8)));
typedef int      v4i __attribute__((ext_vector_type(4)));
typedef float    f4  __attribute__((ext_vector_type(4)));

#if __has_builtin(__builtin_amdgcn_tensor_load_to_lds) && __has_builtin(__builtin_amdgcn_s_wait_tensorcnt)
#define USE_TDM 1
#else
#define USE_TDM 0
#endif

#if USE_TDM
// Issue one TDM descriptor: 2D tile, TROWS rows x D_DIM f32 columns,
// row stride = D_DIM elements.  D# layout per CDNA5 ISA 8.3/8.4.
// 6-arg builtin on this toolchain:
//   (uint32x4 g0, int32x8 g1, int32x4 g2, int32x4 g3, int32x8 aux, i32 cpol)
__device__ __forceinline__ void tdm_issue(uint32_t lds_addr, uint64_t gaddr) {
  v4u g0;
  g0.x = 1u;                                   // count=1 (valid), user mode
  g0.y = lds_addr;                             // LDS byte address
  g0.z = (uint32_t)gaddr;                      // global_addr[31:0]
  g0.w = (uint32_t)((gaddr >> 32) & 0x01FFFFFFu) | 0x80000000u; // ga[56:32] | type=2
  v8i g1;
  g1[0] = (int)(2u << 16);                     // workgroup_mask=0, data_size=2 (4B)
  g1[1] = (int)((uint32_t)D_DIM << 16);        // tensor_dim0[15:0] @ [63:48]
  g1[2] = (int)((uint32_t)TROWS << 16);        // tensor_dim0 hi=0, tensor_dim1[15:0]=TROWS
  g1[3] = (int)((uint32_t)D_DIM << 16);        // tensor_dim1 hi=0, tile_dim0=D_DIM
  g1[4] = (int)TROWS;                          // tile_dim1=TROWS, tile_dim2=0
  g1[5] = (int)D_DIM;                          // tensor_dim0_stride lo (elements)
  g1[6] = 0;                                   // stride hi = 0, dim1_stride lo = 0
  g1[7] = 0;
  v4i gz4  = {0, 0, 0, 0};                     // groups 2/3 unused (2D tensor)
  v8i gz8  = {0, 0, 0, 0, 0, 0, 0, 0};
  __builtin_amdgcn_tensor_load_to_lds(g0, g1, gz4, gz4, gz8, 0);
}
#endif

__global__ __launch_bounds__(THREADS)
void lif_scan_kernel(const float* __restrict__ in,
                     const float* __restrict__ rou,
                     const float* __restrict__ a,
                     const float* __restrict__ b,
                     float* __restrict__ out) {
#if USE_TDM
  __shared__ float tile[2][TROWS * D_DIM];     // 2 x 32 KB double buffer
#endif
  const int tid   = threadIdx.x;
  const int d4    = tid * CH;
  const int batch = blockIdx.y;
  const int t0    = blockIdx.x * TCHUNK;
  const int64_t row0 = (int64_t)batch * T_DIM + t0;

#if USE_TDM
  // Kick off the first tile's DMA immediately: it overlaps the per-channel
  // coefficient loads and the 8 halo rows below.  Wave-uniform scalar branch
  // (readfirstlane) so only wave 0 drives the TDM -- tensor ops ignore EXEC,
  // a lane-masked branch would issue 8 duplicate DMAs.
  const uint64_t gbase = (uint64_t)(uintptr_t)in + (uint64_t)row0 * (D_DIM * 4ull);
  const uint32_t lds0  = (uint32_t)(uintptr_t)&tile[0][0];
  const uint32_t lds1  = (uint32_t)(uintptr_t)&tile[1][0];
  const bool is_wave0 = (__builtin_amdgcn_readfirstlane(threadIdx.x) == 0);
  if (is_wave0) tdm_issue(lds0, gbase);        // prologue: tile 0 -> buf 0
#endif

  // per-channel coefficients
  f4 r  = *(const f4*)(rou + d4);
  f4 av = *(const f4*)(a + d4);
  f4 bv = *(const f4*)(b + d4);
  f4 rb = r * bv;                // prefix = r - rb*x
  f4 cc = (1.0f - r) * av;       // drive  = cc*x

  float run[K_TRUNC + 1][CH];
#pragma unroll
  for (int k = 0; k <= K_TRUNC; ++k)
#pragma unroll
    for (int c = 0; c < CH; ++c) run[k][c] = 0.0f;

  // advance recurrence one time step; returns out[t]
  auto advance = [&](f4 x) -> f4 {
    f4 pre = r - rb * x;
    f4 drv = cc * x;
#pragma unroll
    for (int k = K_TRUNC; k >= 1; --k)
#pragma unroll
      for (int c = 0; c < CH; ++c) run[k][c] = run[k - 1][c] * pre[c];
#pragma unroll
    for (int c = 0; c < CH; ++c) run[0][c] = drv[c];
    f4 s = drv;
#pragma unroll
    for (int k = 1; k <= K_TRUNC; ++k)
#pragma unroll
      for (int c = 0; c < CH; ++c) s[c] += run[k][c];
    return s;
  };

  // ---- warm-up: 8 halo rows straight from global (absent at batch start) ----
  if (t0 > 0) {
#pragma unroll
    for (int w = 0; w < K_TRUNC; ++w) {
      f4 x = __builtin_nontemporal_load(
          (const f4*)(in + (row0 - K_TRUNC + w) * D_DIM + d4));
      (void)advance(x);   // no store
    }
  }

#if USE_TDM
  for (int i = 0; i < NTILES; ++i) {
    if (is_wave0) {
      if (i + 1 < NTILES) {
        tdm_issue(((i + 1) & 1) ? lds1 : lds0,
                  gbase + (uint64_t)(i + 1) * (TROWS * D_DIM * 4ull));
        __builtin_amdgcn_s_wait_tensorcnt(1);  // tile i done, i+1 in flight
      } else {
        __builtin_amdgcn_s_wait_tensorcnt(0);
      }
    }
    __syncthreads();                           // tile i visible to all waves
    const float* src = (i & 1) ? &tile[1][0] : &tile[0][0];
    float* outp = out + (row0 + (int64_t)i * TROWS) * D_DIM + d4;
#pragma unroll
    for (int j = 0; j < TROWS; ++j) {
      f4 x = *(const f4*)(src + j * D_DIM + d4);
      f4 s = advance(x);
      __builtin_nontemporal_store(s, (f4*)(outp + (size_t)j * D_DIM));
    }
    __syncthreads();                           // buffer free before tile i+2
  }
#else
  // fallback: direct streaming (still coalesced b128 loads/stores)
  const float* src = in + row0 * D_DIM + d4;
  float* outp = out + row0 * D_DIM + d4;
  for (int t = 0; t < TCHUNK; ++t) {
    f4 x = __builtin_nontemporal_load((const f4*)(src + (size_t)t * D_DIM));
    f4 s = advance(x);
    __builtin_nontemporal_store(s, (f4*)(outp + (size_t)t * D_DIM));
  }
#endif
}

extern "C" void kernel_launch(void* const* d_in, const int* in_sizes, int n_in,
                              void* d_out, int out_size, void* d_ws, size_t ws_size,
                              hipStream_t stream) {
  (void)in_sizes; (void)n_in; (void)out_size; (void)d_ws; (void)ws_size;
  const float* in  = (const float*)d_in[0];
  const float* rou = (const float*)d_in[1];
  const float* a   = (const float*)d_in[2];
  const float* b   = (const float*)d_in[3];
  float* out = (float*)d_out;
  dim3 grid(T_DIM / TCHUNK, B_DIM);            // 32 x 8 = 256 workgroups
  lif_scan_kernel<<<grid, THREADS, 0, stream>>>(in, rou, a, b, out);
}